// BatchLPCSynth_75608604278812
// MI455X (gfx1250) — compile-verified
//
#include <hip/hip_runtime.h>
#include <stdint.h>

// Problem constants (from reference)
#define HOP   256
#define WIN   1024
#define PADL  384                 // (WIN - HOP) / 2
#define BATCH 16
#define TLEN  262144              // 2^18
#define NFRM  1024                // TLEN / HOP, frames per batch row
#define NP    22                  // LPC order

// Staging geometry: one wave32 workgroup handles 32 consecutive frames of one row.
#define NF_PER_WG    32
#define GRPS_PER_ROW (NFRM / NF_PER_WG)            // 32
#define SPAN         ((NF_PER_WG - 1) * HOP + WIN) // 8960 floats staged per WG
#define LDS_FLOATS   (SPAN + WIN)                  // + Hann table
#define LDS_BYTES    (LDS_FLOATS * 4)              // 39,936 B (< 64KB default cap)

// Workspace layout (floats): [0,16M) windowed frames [b][f][t]; then Hann
// table (1024); then reciprocal OLA norm (TLEN).
#define FRAMES_FLOATS (BATCH * NFRM * WIN)         // 16,777,216
#define WINTAB_OFF    FRAMES_FLOATS
#define INORM_OFF     (FRAMES_FLOATS + WIN)

typedef unsigned int u32x4 __attribute__((ext_vector_type(4)));
typedef int          i32x4 __attribute__((ext_vector_type(4)));
typedef int          i32x8 __attribute__((ext_vector_type(8)));

#if defined(__has_builtin)
#if __has_builtin(__builtin_amdgcn_tensor_load_to_lds)
#define USE_TDM 1
#endif
#endif

// ---------------------------------------------------------------------------
// Tensor Data Mover: 1D row DMA  global -> LDS.
//  tile_elems  : f32 elements to transfer (tile_dim0)
//  valid_elems : elements remaining in the source row (tensor_dim0);
//                reads past this return zero (free right-edge padding).
// Descriptor packing per CDNA5 ISA ch.8 (D# group0/group1).
// ---------------------------------------------------------------------------
__device__ __forceinline__ void stage_tdm(uint32_t lds_byte, const float* gsrc,
                                          uint32_t tile_elems, uint32_t valid_elems) {
#ifdef USE_TDM
  uint64_t ga = (uint64_t)(size_t)gsrc;
  u32x4 g0;
  g0.x = 1u;                                                // count=1, user descriptor
  g0.y = lds_byte;                                          // lds_addr (bytes)
  g0.z = (uint32_t)ga;                                      // global_addr[31:0]
  g0.w = (uint32_t)((ga >> 32) & 0x01FFFFFFu) | (2u << 30); // addr[56:32] | type=2
  i32x8 g1;
  g1[0] = 0x00020000;                                       // data_size = 4 bytes
  g1[1] = (int)((valid_elems & 0xFFFFu) << 16);             // tensor_dim0[15:0]
  g1[2] = (int)((valid_elems >> 16) & 0xFFFFu) | (1 << 16); // tensor_dim0[31:16], tensor_dim1=1
  g1[3] = (int)((tile_elems & 0xFFFFu) << 16);              // tile_dim0
  g1[4] = 0;                                                // tile_dim1=0 (1D), tile_dim2=0
  g1[5] = (int)valid_elems;                                 // tensor_dim0_stride (unused, 1D)
  g1[6] = 0;
  g1[7] = 0;
  i32x4 z4 = {0, 0, 0, 0};                                  // groups 2/3 disabled (<=2D)
#if __clang_major__ >= 23
  i32x8 z8 = {0, 0, 0, 0, 0, 0, 0, 0};
  __builtin_amdgcn_tensor_load_to_lds(g0, g1, z4, z4, z8, 0);
#else
  __builtin_amdgcn_tensor_load_to_lds(g0, g1, z4, z4, 0);
#endif
#endif
}

__device__ __forceinline__ void wait_tensorcnt0() {
#ifdef USE_TDM
#if __has_builtin(__builtin_amdgcn_s_wait_tensorcnt)
  __builtin_amdgcn_s_wait_tensorcnt(0);
#else
  asm volatile("s_wait_tensorcnt 0" ::: "memory");
#endif
#endif
}

__device__ __forceinline__ float hann_w(int n) {
  const float k = 6.135923151542565e-03f;  // 2*pi / WIN
  return 0.5f - 0.5f * cosf(k * (float)n);
}

// ---------------------------------------------------------------------------
// Kernel 0: one-time tables — Hann window (1024) and reciprocal OLA norm
// (TLEN entries; depends on t only, shared by all batch rows).
// ---------------------------------------------------------------------------
__global__ __launch_bounds__(256) void prep_kernel(float* __restrict__ ws) {
  const int t = blockIdx.x * blockDim.x + threadIdx.x;
  if (t < WIN) ws[WINTAB_OFF + t] = hann_w(t);
  if (t < TLEN) {
    const int tp = t + PADL;
    int f_hi = tp >> 8;                 if (f_hi > NFRM - 1) f_hi = NFRM - 1;
    int f_lo = (tp - (WIN - HOP)) >> 8; if (f_lo < 0) f_lo = 0;
    float den = 0.0f;
    for (int f = f_lo; f <= f_hi; ++f) den += hann_w(tp - (f << 8));
    ws[INORM_OFF + t] = 1.0f / den;
  }
}

// ---------------------------------------------------------------------------
// One LPC synthesis step:  y = g*x - sum_p A[p]*s[p];  s <- [y, s[0..NP-2]]
// Two accumulation chains halve the FMA dependency depth. The shift copies are
// pure SSA moves erased by the compiler inside 44-sample unrolled blocks
// (44 mod 22 == 0, so the net rotation per block is the identity).
// ---------------------------------------------------------------------------
__device__ __forceinline__ float lpc_step(float x, float g,
                                          const float A[NP], float s[NP]) {
  float acc0 = A[0] * s[0];
  float acc1 = A[1] * s[1];
#pragma unroll
  for (int p = 2; p < NP; p += 2) {
    acc0 = fmaf(A[p],     s[p],     acc0);
    acc1 = fmaf(A[p + 1], s[p + 1], acc1);
  }
  float y = fmaf(g, x, -(acc0 + acc1));
#pragma unroll
  for (int p = NP - 1; p > 0; --p) s[p] = s[p - 1];
  s[0] = y;
  return y;
}

// ---------------------------------------------------------------------------
// Kernel 1: per-frame order-22 IIR over its 1024-sample window.
// One wave32 workgroup = 32 consecutive frames of one batch row. Two TDM
// loads stage (a) the shared excitation span and (b) the Hann table into LDS;
// same-wave tensor ops are in-order, one s_wait_tensorcnt 0 covers both.
// Windowed frames (y * w[t]) go to workspace laid out [b][f][t].
// ---------------------------------------------------------------------------
__global__ __launch_bounds__(32) void lpc_iir_kernel(
    const float* __restrict__ ex, const float* __restrict__ gain,
    const float* __restrict__ acoef, float* __restrict__ ws) {
  extern __shared__ float smem[];
  const int wg  = blockIdx.x;
  const int b   = wg / GRPS_PER_ROW;
  const int fg  = wg - b * GRPS_PER_ROW;
  const int f0  = fg * NF_PER_WG;
  const int tid = threadIdx.x;

  const size_t row0   = (size_t)b * TLEN;
  const int start     = f0 * HOP - PADL;          // position in un-padded row coords
  const int g_start   = start < 0 ? 0 : start;
  const int lds_skip  = start < 0 ? -start : 0;   // left zero-pad (first group only)

  if (lds_skip > 0) {
    for (int i = tid; i < lds_skip; i += 32) smem[i] = 0.0f;
  }

#ifdef USE_TDM
  uint32_t lds_base = (uint32_t)(size_t)(void*)&smem[0];
  stage_tdm(lds_base + 4u * (uint32_t)lds_skip, ex + row0 + g_start,
            (uint32_t)(SPAN - lds_skip), (uint32_t)(TLEN - g_start));
  stage_tdm(lds_base + 4u * (uint32_t)SPAN, ws + WINTAB_OFF, WIN, WIN);
  wait_tensorcnt0();
  asm volatile("" ::: "memory");
#else
  const int n = SPAN - lds_skip;
  for (int i = tid; i < n; i += 32) {
    int gi = g_start + i;
    smem[lds_skip + i] = (gi < TLEN) ? ex[row0 + gi] : 0.0f;
  }
  for (int i = tid; i < WIN; i += 32) smem[SPAN + i] = ws[WINTAB_OFF + i];
  __syncthreads();
#endif

  const int frame = b * NFRM + f0 + tid;
  const float g = gain[frame];
  float A[NP], s[NP];
#pragma unroll
  for (int p = 0; p < NP; ++p) A[p] = acoef[(size_t)frame * NP + p];
#pragma unroll
  for (int p = 0; p < NP; ++p) s[p] = 0.0f;

  const float4* __restrict__ xin  = (const float4*)(smem + tid * HOP);
  const float4* __restrict__ wtab = (const float4*)(smem + SPAN);  // broadcast reads
  float4* __restrict__       yout = (float4*)(ws + (size_t)frame * WIN);

  // 23 blocks x 44 samples = 1012, then a 12-sample tail (1012 = 46*22, so the
  // tail also starts at rotation phase 0).
  int q = 0;  // float4 chunk index within the frame
  for (int blk = 0; blk < 23; ++blk) {
#pragma unroll
    for (int c = 0; c < 11; ++c) {
      float4 x4 = xin[q + c];
      float4 w4 = wtab[q + c];
      float4 y4;
      y4.x = lpc_step(x4.x, g, A, s) * w4.x;
      y4.y = lpc_step(x4.y, g, A, s) * w4.y;
      y4.z = lpc_step(x4.z, g, A, s) * w4.z;
      y4.w = lpc_step(x4.w, g, A, s) * w4.w;
      yout[q + c] = y4;
    }
    q += 11;
  }
#pragma unroll
  for (int c = 0; c < 3; ++c) {
    float4 x4 = xin[253 + c];
    float4 w4 = wtab[253 + c];
    float4 y4;
    y4.x = lpc_step(x4.x, g, A, s) * w4.x;
    y4.y = lpc_step(x4.y, g, A, s) * w4.y;
    y4.z = lpc_step(x4.z, g, A, s) * w4.z;
    y4.w = lpc_step(x4.w, g, A, s) * w4.w;
    yout[253 + c] = y4;
  }
}

// ---------------------------------------------------------------------------
// Kernel 2: overlap-add of the pre-windowed frames + precomputed reciprocal
// norm. Pure memory: <=4 coalesced loads + 1 table load + 3 adds + 1 mul.
// Deterministic gather form, no atomics. Frame set matches the reference crop.
// ---------------------------------------------------------------------------
__global__ __launch_bounds__(256) void ola_kernel(const float* __restrict__ ws,
                                                  float* __restrict__ out) {
  const int idx = blockIdx.x * blockDim.x + threadIdx.x;
  if (idx >= BATCH * TLEN) return;
  const int b  = idx >> 18;            // TLEN == 2^18
  const int t  = idx & (TLEN - 1);
  const int tp = t + PADL;             // position in padded/OLA coords

  int f_hi = tp >> 8;                  // HOP == 256
  if (f_hi > NFRM - 1) f_hi = NFRM - 1;
  int f_lo = (tp - (WIN - HOP)) >> 8;  // floor((tp-768)/256) == ceil((tp-1023)/256)
  if (f_lo < 0) f_lo = 0;

  float num = 0.0f;
  for (int f = f_lo; f <= f_hi; ++f) {
    int off = tp - (f << 8);
    num += ws[(((size_t)b << 10) + (size_t)f) * WIN + off];
  }
  out[idx] = num * ws[INORM_OFF + t];
}

// ---------------------------------------------------------------------------
// Entry point. Inputs: ex (B*T f32), gain (B*F f32), a (B*F*P f32).
// Output: (B*T f32). Workspace: ~65.1 MB of d_ws (frames + tables).
// ---------------------------------------------------------------------------
extern "C" void kernel_launch(void* const* d_in, const int* in_sizes, int n_in,
                              void* d_out, int out_size, void* d_ws, size_t ws_size,
                              hipStream_t stream) {
  (void)in_sizes; (void)n_in; (void)out_size; (void)ws_size;
  const float* ex    = (const float*)d_in[0];
  const float* gain  = (const float*)d_in[1];
  const float* acoef = (const float*)d_in[2];
  float* out = (float*)d_out;
  float* ws  = (float*)d_ws;   // needs (FRAMES_FLOATS + WIN + TLEN) * 4 bytes

  prep_kernel<<<TLEN / 256, 256, 0, stream>>>(ws);
  lpc_iir_kernel<<<BATCH * GRPS_PER_ROW, 32, LDS_BYTES, stream>>>(ex, gain, acoef, ws);
  ola_kernel<<<(BATCH * TLEN + 255) / 256, 256, 0, stream>>>(ws, out);
}